// GNN_82781199663565
// MI455X (gfx1250) — compile-verified
//
#include <hip/hip_runtime.h>
#include <hip/hip_bf16.h>

typedef __attribute__((ext_vector_type(2))) float v2f;
typedef __attribute__((ext_vector_type(8))) float v8f;

// Problem constants
#define HW      65536      // 256*256
#define NPIX    131072     // B*HW
#define KDIM    256        // input channels
#define OCH     60         // 40 (upper) + 20 (lower) conv output channels
#define OPAD    64
#define LSTRIDE 264        // LDS row stride (floats), padded vs 256 to avoid bank conflicts

// ---------------- kernel 1: zero the stats accumulators ----------------
__global__ void pg_init_stats(float* __restrict__ stats) {
    if (threadIdx.x < 128) stats[threadIdx.x] = 0.f;
}

// ---------------- kernel 2: fused GEMM (WMMA f32 16x16x4) + attention scale + BN stats ----------------
// grid = 512 workgroups, 256 threads (8 waves): wave = mt(4 M-tiles of 16 ch) x pg(2 pixel groups of 128)
// x tiles stream into LDS via async-to-LDS loads (ASYNCcnt), double buffered.
__global__ __launch_bounds__(256) void pg_gemm_stats(
    const float* __restrict__ xp,      // [B,256,H,W]
    const float* __restrict__ h_att,   // [3,B,1,H,W]
    const float* __restrict__ Wu,      // [40,256]
    const float* __restrict__ Wl,      // [20,256]
    float* __restrict__ stats,         // [2][64]  (sum, sumsq)
    float* __restrict__ y)             // [60][NPIX]
{
    __shared__ __align__(16) float xbuf[2][16 * LSTRIDE];
    __shared__ float ssum[OPAD];
    __shared__ float ssq[OPAD];

    const int t    = threadIdx.x;
    const int wave = t >> 5;
    const int lane = t & 31;
    const int ln   = lane & 15;
    const bool hi  = (lane >= 16);
    const int mt   = wave & 3;   // M tile (16 channels)
    const int pg   = wave >> 2;  // pixel group (128 pixels)

    const int wg    = blockIdx.x;
    const int b     = wg >> 8;            // 256 workgroups per batch image
    const int sbase = (wg & 255) * 256;   // spatial base within image
    const int pixoff = pg * 128;

    if (t < OPAD) { ssum[t] = 0.f; ssq[t] = 0.f; }

    // A-fragment source: row m of fused weight matrix [Wu;Wl;0-pad]
    const int  m      = mt * 16 + ln;
    const bool wvalid = (m < OCH);
    const float* wrow = (m < 40) ? (Wu + (size_t)m * KDIM)
                      : (m < 60) ? (Wl + (size_t)(m - 40) * KDIM)
                                 : Wu;   // safe dummy, value masked to 0
    const int koff = hi ? 2 : 0;         // lanes 16-31 hold K+2,K+3 per ISA A layout

    const float* xb = xp + (size_t)b * (KDIM * HW) + sbase;

    // per-lane async staging geometry: 4 x b128 per lane per 16x256 chunk
    const int r0l = t >> 6;          // base row (0..3), +4 per issue
    const int cl  = (t * 4) & 255;   // float column (16B aligned)

    v8f acc[8];
#pragma unroll
    for (int i = 0; i < 8; ++i) acc[i] = v8f{0.f,0.f,0.f,0.f,0.f,0.f,0.f,0.f};

    // ---- issue chunk 0 into buffer 0 (async global -> LDS, no VGPR round trip) ----
    {
#pragma unroll
        for (int i = 0; i < 4; ++i) {
            const int r = r0l + 4 * i;
            const unsigned voff  = (unsigned)(((0 + r) * HW + cl) * 4);
            const unsigned laddr = (unsigned)(uintptr_t)&xbuf[0][r * LSTRIDE + cl];
            asm volatile("global_load_async_to_lds_b128 %0, %1, %2"
                         :: "v"(laddr), "v"(voff), "s"(xb) : "memory");
        }
    }

    for (int chunk = 0; chunk < 16; ++chunk) {
        const int k0  = chunk * 16;
        const int cur = chunk & 1;
        // my async loads for this chunk have landed in LDS...
        asm volatile("s_wait_asynccnt 0x0" ::: "memory");
        // ...and after the barrier everyone's have.
        __syncthreads();
        // prefetch next chunk into the other buffer (last read before this barrier -> safe)
        if (chunk + 1 < 16) {
#pragma unroll
            for (int i = 0; i < 4; ++i) {
                const int r = r0l + 4 * i;
                const unsigned voff  = (unsigned)(((k0 + 16 + r) * HW + cl) * 4);
                const unsigned laddr = (unsigned)(uintptr_t)&xbuf[1 - cur][r * LSTRIDE + cl];
                asm volatile("global_load_async_to_lds_b128 %0, %1, %2"
                             :: "v"(laddr), "v"(voff), "s"(xb) : "memory");
            }
        }
        const float* __restrict__ xs = &xbuf[cur][0];
#pragma unroll
        for (int ks = 0; ks < 4; ++ks) {
            const int k = k0 + ks * 4;
            // A 16x4 f32: lanes 0-15 -> {K,K+1}, lanes 16-31 -> {K+2,K+3}
            const float2 araw = *(const float2*)(wrow + k + koff);
            v2f a;
            a.x = wvalid ? araw.x : 0.f;
            a.y = wvalid ? araw.y : 0.f;
            const int r0 = ks * 4 + koff;
#pragma unroll
            for (int nt = 0; nt < 8; ++nt) {
                const int col = pixoff + nt * 16 + ln;
                // B 4x16 f32 mirrored layout: lanes 0-15 rows {K,K+1}, lanes 16-31 rows {K+2,K+3}
                v2f bb;
                bb.x = xs[r0 * LSTRIDE + col];
                bb.y = xs[(r0 + 1) * LSTRIDE + col];
                acc[nt] = __builtin_amdgcn_wmma_f32_16x16x4_f32(
                    false, a, false, bb, (short)0, acc[nt], false, false);
            }
        }
    }

    // ---- epilogue: attention scale, store y, accumulate BN sums ----
    float vsum[8], vsq[8];
#pragma unroll
    for (int r = 0; r < 8; ++r) { vsum[r] = 0.f; vsq[r] = 0.f; }

#pragma unroll
    for (int nt = 0; nt < 8; ++nt) {
        const int sp  = sbase + pixoff + nt * 16 + ln;     // spatial index in image
        const int pix = b * HW + sp;                       // global pixel index
        const float a1 = h_att[(size_t)(1 * 2 + b) * HW + sp];
        const float a2 = h_att[(size_t)(2 * 2 + b) * HW + sp];
#pragma unroll
        for (int r = 0; r < 8; ++r) {
            // C/D layout: lanes 0-15 -> M=r, lanes 16-31 -> M=r+8
            const int o = mt * 16 + r + (hi ? 8 : 0);
            const float att = (o < 40) ? a1 : a2;
            const float val = acc[nt][r] * att;
            if (o < OCH) y[(size_t)o * NPIX + pix] = val;
            vsum[r] += val;
            vsq[r]  += val * val;
        }
    }

    // reduce each row over its 16 lanes (halves are independent channels)
#pragma unroll
    for (int r = 0; r < 8; ++r) {
        float s = vsum[r], q = vsq[r];
#pragma unroll
        for (int d = 1; d < 16; d <<= 1) {
            s += __shfl_xor(s, d, 16);
            q += __shfl_xor(q, d, 16);
        }
        if (ln == 0) {
            const int o = mt * 16 + r + (hi ? 8 : 0);
            atomicAdd(&ssum[o], s);
            atomicAdd(&ssq[o], q);
        }
    }
    __syncthreads();
    if (t < 128) {
        const int ch = t & 63;
        const int st = t >> 6;
        if (ch < OCH)
            atomicAdd(&stats[st * OPAD + ch], st ? ssq[ch] : ssum[ch]);
    }
}

// ---------------- kernel 3: finalize BN -> fused scale/shift ----------------
__global__ void pg_finalize(const float* __restrict__ stats,
                            const float* __restrict__ gu, const float* __restrict__ bu,
                            const float* __restrict__ gl, const float* __restrict__ bl,
                            float* __restrict__ scsh) {   // [2][64]
    const int t = threadIdx.x;
    if (t < OCH) {
        const float n    = (float)NPIX;
        const float mean = stats[t] / n;
        const float var  = stats[OPAD + t] / n - mean * mean;
        const float rstd = rsqrtf(var + 1e-5f);
        const float g  = (t < 40) ? gu[t] : gl[t - 40];
        const float be = (t < 40) ? bu[t] : bl[t - 40];
        const float sc = g * rstd;
        scsh[t]        = sc;
        scsh[OPAD + t] = be - mean * sc;
    }
}

// ---------------- kernel 4: normalize + ReLU + combine with p_nodes/h_nodes ----------------
// out[7,B,HID,H,W]; float4 per thread; 8960 blocks x 256 threads covers 9175040 elements exactly
__global__ __launch_bounds__(256) void pg_combine(
    const float* __restrict__ p_nodes,   // [7,B,10,H,W]
    const float* __restrict__ h_nodes,   // [3,B,10,H,W]
    const float* __restrict__ y,         // [60][NPIX]
    const float* __restrict__ scsh,      // [2][64]
    float* __restrict__ out)
{
    const unsigned g = blockIdx.x * 256u + threadIdx.x;
    const unsigned e = g * 4u;                    // flat output element base
    const unsigned part = e / 1310720u;           // 2*10*65536
    unsigned rem = e - part * 1310720u;
    const unsigned b   = rem / 655360u;           // 10*65536
    rem -= b * 655360u;
    const unsigned hid = rem >> 16;
    const unsigned s   = rem & 65535u;

    const size_t pidx = ((size_t)(part * 2 + b) * 10 + hid) * HW + s;
    const float4 pv = *(const float4*)(p_nodes + pidx);
    float4 ov;
    if (part == 0) {
        const size_t hidx = ((size_t)b * 10 + hid) * HW + s;
        const float4 hv = *(const float4*)(h_nodes + hidx);
        ov.x = 0.5f * (pv.x + hv.x);
        ov.y = 0.5f * (pv.y + hv.y);
        ov.z = 0.5f * (pv.z + hv.z);
        ov.w = 0.5f * (pv.w + hv.w);
    } else {
        const int o = (part <= 4) ? (int)((part - 1) * 10 + hid)
                                  : (int)(40 + (part - 5) * 10 + hid);
        const float sc = scsh[o];
        const float sh = scsh[OPAD + o];
        const float4 yv = *(const float4*)(y + (size_t)o * NPIX + (size_t)b * HW + s);
        const float r0 = fmaxf(fmaf(yv.x, sc, sh), 0.f);
        const float r1 = fmaxf(fmaf(yv.y, sc, sh), 0.f);
        const float r2 = fmaxf(fmaf(yv.z, sc, sh), 0.f);
        const float r3 = fmaxf(fmaf(yv.w, sc, sh), 0.f);
        ov.x = 0.5f * (pv.x + r0);
        ov.y = 0.5f * (pv.y + r1);
        ov.z = 0.5f * (pv.z + r2);
        ov.w = 0.5f * (pv.w + r3);
    }
    *(float4*)(out + e) = ov;
}

// ---------------- launch ----------------
extern "C" void kernel_launch(void* const* d_in, const int* in_sizes, int n_in,
                              void* d_out, int out_size, void* d_ws, size_t ws_size,
                              hipStream_t stream) {
    const float* p_nodes = (const float*)d_in[0];
    const float* h_nodes = (const float*)d_in[1];
    // d_in[2] = f_nodes (unused by reference output)
    const float* xp      = (const float*)d_in[3];
    const float* h_att   = (const float*)d_in[4];
    const float* Wu      = (const float*)d_in[5];
    const float* Wl      = (const float*)d_in[6];
    const float* gamma_u = (const float*)d_in[7];
    const float* beta_u  = (const float*)d_in[8];
    const float* gamma_l = (const float*)d_in[9];
    const float* beta_l  = (const float*)d_in[10];

    float* ws    = (float*)d_ws;
    float* stats = ws;          // [2][64]
    float* scsh  = ws + 128;    // [2][64]
    float* ybuf  = ws + 256;    // [60][NPIX] = 31.5 MB

    pg_init_stats<<<1, 128, 0, stream>>>(stats);
    pg_gemm_stats<<<512, 256, 0, stream>>>(xp, h_att, Wu, Wl, stats, ybuf);
    pg_finalize<<<1, 64, 0, stream>>>(stats, gamma_u, beta_u, gamma_l, beta_l, scsh);
    pg_combine<<<8960, 256, 0, stream>>>(p_nodes, h_nodes, ybuf, scsh, (float*)d_out);
}